// FPS_89223650607123
// MI455X (gfx1250) — compile-verified
//
#include <hip/hip_runtime.h>
#include <cstdint>
#include <cstddef>

// Farthest Point Sampling for MI455X (gfx1250, wave32).
//   points: B x N x 3 f32  ->  out: B x S x 3 f32
// One workgroup per batch. Running min-dist array lives in global workspace
// (L2-resident), vectorized b128 traffic. Winner-point broadcast each
// iteration uses CDNA5 async global->LDS DMA + s_wait_asynccnt.

#define FPS_N   131072
#define FPS_S   1024
#define FPS_T   1024                  // threads per block (32 wave32s)
#define FPS_G   (FPS_N / (FPS_T * 4)) // 32 groups of 4 points per thread
#define FPS_NW  (FPS_T / 32)          // 32 waves
#define FPS_BIG 1e10f

typedef __attribute__((address_space(1))) int fps_gint;  // global int*
typedef __attribute__((address_space(3))) int fps_lint;  // LDS int*

// ---- CDNA5 async global->LDS load of 4 bytes (one lane) ------------------
__device__ __forceinline__ void fps_async_ld_b32(const float* g, float* l) {
#if __has_builtin(__builtin_amdgcn_global_load_async_to_lds_b32)
  __builtin_amdgcn_global_load_async_to_lds_b32(
      (fps_gint*)g, (fps_lint*)l, 0 /*offset*/, 0 /*cpol*/);
#else
  unsigned lds_off = (unsigned)(size_t)l;   // low 32 bits of generic shared ptr = LDS offset
  asm volatile("global_load_async_to_lds_b32 %0, %1, off"
               :: "v"(lds_off), "v"(g) : "memory");
#endif
}

__device__ __forceinline__ void fps_wait_async0() {
#if __has_builtin(__builtin_amdgcn_s_wait_asynccnt)
  __builtin_amdgcn_s_wait_asynccnt(0);
#else
  asm volatile("s_wait_asynccnt 0" ::: "memory");
#endif
}

template <bool USE_WS>
__global__ __launch_bounds__(FPS_T) void FPS_89223650607123_kernel(
    const float* __restrict__ pts, float* __restrict__ out,
    float* __restrict__ dws) {
  __shared__ float s_q[4];           // broadcast coords of current query point
  __shared__ float s_rv[FPS_NW];     // per-wave best value
  __shared__ int   s_ri[FPS_NW];     // per-wave best index
  __shared__ int   s_widx;           // workgroup winner index

  const int tid = threadIdx.x;
  const int b   = blockIdx.x;
  const float4* __restrict__ pb4 =
      (const float4*)(pts + (size_t)b * FPS_N * 3);   // 16B-aligned per batch
  float* __restrict__ obase = out + (size_t)b * FPS_S * 3;
  float4* __restrict__ db4 =
      USE_WS ? (float4*)(dws + (size_t)b * FPS_N) : (float4*)nullptr;

  // Fallback only: per-thread dist in scratch (4 points per group).
  float4 distloc[USE_WS ? 1 : FPS_G];

  // First selected point is index 0 (reference init).
  if (tid < 3) s_q[tid] = ((const float*)pb4)[tid];
  __syncthreads();

  for (int s = 0; s < FPS_S; ++s) {
    const float qx = s_q[0];
    const float qy = s_q[1];
    const float qz = s_q[2];
    // Emit current query point as sample s (reference emits before update).
    if (tid < 3) obase[s * 3 + tid] = s_q[tid];

    float bestv = -1.0f;
    int   besti = 0x7fffffff;
    const bool first = (s == 0);    // min(1e10, d) == d  -> no dist read/init

    for (int g = 0; g < FPS_G; ++g) {
      const int pi = 3 * (g * FPS_T + tid);   // float4 index: 48B/lane, dense
      const float4 A = pb4[pi + 0];           // p0.xyz  p1.x
      const float4 Bv = pb4[pi + 1];          // p1.yz   p2.xy
      const float4 C = pb4[pi + 2];           // p2.z    p3.xyz

      float dx, dy, dz;
      dx = A.x - qx;  dy = A.y - qy;  dz = A.z - qz;
      const float d0 = dx * dx + dy * dy + dz * dz;
      dx = A.w - qx;  dy = Bv.x - qy; dz = Bv.y - qz;
      const float d1 = dx * dx + dy * dy + dz * dz;
      dx = Bv.z - qx; dy = Bv.w - qy; dz = C.x - qz;
      const float d2 = dx * dx + dy * dy + dz * dz;
      dx = C.y - qx;  dy = C.z - qy;  dz = C.w - qz;
      const float d3 = dx * dx + dy * dy + dz * dz;

      float4 nd;
      if (first) {
        nd = make_float4(d0, d1, d2, d3);
      } else {
        const float4 old = USE_WS ? db4[g * FPS_T + tid] : distloc[g];
        nd.x = fminf(old.x, d0);
        nd.y = fminf(old.y, d1);
        nd.z = fminf(old.z, d2);
        nd.w = fminf(old.w, d3);
      }
      if (USE_WS) db4[g * FPS_T + tid] = nd; else distloc[g] = nd;

      // idx increases with scan order -> strict '>' keeps first max (argmax).
      const int idx0 = g * (FPS_T * 4) + tid * 4;
      if (nd.x > bestv) { bestv = nd.x; besti = idx0;     }
      if (nd.y > bestv) { bestv = nd.y; besti = idx0 + 1; }
      if (nd.z > bestv) { bestv = nd.z; besti = idx0 + 2; }
      if (nd.w > bestv) { bestv = nd.w; besti = idx0 + 3; }
    }

    // Wave32 reduction (prefer larger value; on exact tie, lower index).
#pragma unroll
    for (int m = 16; m >= 1; m >>= 1) {
      const float ov = __shfl_xor(bestv, m, 32);
      const int   oi = __shfl_xor(besti, m, 32);
      if (ov > bestv || (ov == bestv && oi < besti)) { bestv = ov; besti = oi; }
    }
    if ((tid & 31) == 0) { s_rv[tid >> 5] = bestv; s_ri[tid >> 5] = besti; }
    __syncthreads();

    // Cross-wave reduction in wave 0 (32 candidates in lanes 0..31).
    if (tid < 32) {
      float v = s_rv[tid];
      int   i = s_ri[tid];
#pragma unroll
      for (int m = 16; m >= 1; m >>= 1) {
        const float ov = __shfl_xor(v, m, 32);
        const int   oi = __shfl_xor(i, m, 32);
        if (ov > v || (ov == v && oi < i)) { v = ov; i = oi; }
      }
      if (tid == 0) s_widx = i;
    }
    __syncthreads();

    // Broadcast winner coords to LDS via async global->LDS DMA (3 lanes x 4B).
    const int widx = s_widx;
    if (tid < 3)
      fps_async_ld_b32((const float*)pb4 + (size_t)widx * 3 + tid, &s_q[tid]);
    fps_wait_async0();   // issuing wave drains ASYNCcnt; others pass instantly
    __syncthreads();
  }
}

extern "C" void kernel_launch(void* const* d_in, const int* in_sizes, int n_in,
                              void* d_out, int out_size, void* d_ws, size_t ws_size,
                              hipStream_t stream) {
  (void)n_in; (void)out_size;
  const float* pts = (const float*)d_in[0];
  float* out       = (float*)d_out;
  const int B = in_sizes[0] / (FPS_N * 3);   // 32 for the reference shapes
  const size_t need = (size_t)B * FPS_N * sizeof(float);
  if (d_ws != nullptr && ws_size >= need) {
    FPS_89223650607123_kernel<true><<<dim3(B), dim3(FPS_T), 0, stream>>>(
        pts, out, (float*)d_ws);
  } else {
    FPS_89223650607123_kernel<false><<<dim3(B), dim3(FPS_T), 0, stream>>>(
        pts, out, nullptr);
  }
}